// AdvLossForTarget_max_1606317768943
// MI455X (gfx1250) — compile-verified
//
#include <hip/hip_runtime.h>
#include <hip/hip_bf16.h>
#include <math.h>

// Problem constants (match reference).
constexpr int   Bsz  = 262144;
constexpr int   Csz  = 1001;
constexpr float EPSC = 1e-6f;

constexpr int WAVE            = 32;   // gfx1250 is wave32-only
constexpr int WAVES_PER_BLOCK = 8;    // 256 threads
constexpr int ROWS_PER_WAVE   = 8;
constexpr int THREADS         = WAVE * WAVES_PER_BLOCK;
constexpr int ROWS_PER_BLOCK  = WAVES_PER_BLOCK * ROWS_PER_WAVE;  // 64
constexpr int GRID1           = Bsz / ROWS_PER_BLOCK;             // 4096

constexpr int FULL_ITERS = (Csz - 1) / WAVE;          // 31 -> columns [0, 992)
constexpr int TAIL_BASE  = FULL_ITERS * WAVE;         // 992
constexpr int TAIL_LANES = Csz - TAIL_BASE;           // 9  -> columns 992..1000
constexpr int LAST_LANE  = (Csz - 1) & (WAVE - 1);    // 8  (holds column 1000)

// Online softmax update: one expf on the (common) no-new-max path.
__device__ __forceinline__ void online_update(float x, float& m, float& s) {
    if (x > m) {
        s = s * expf(m - x) + 1.0f;   // rare after warm-up
        m = x;
    } else {
        s += expf(x - m);             // common path: single transcendental
    }
}

// Stage 1: one wave32 per row, single-pass online softmax, per-block partial sum.
__global__ __launch_bounds__(THREADS)
void row_loss_kernel(const float* __restrict__ inp, float* __restrict__ partial) {
    const int lane = threadIdx.x & (WAVE - 1);
    const int wid  = threadIdx.x >> 5;
    const long long waveGlobal = (long long)blockIdx.x * WAVES_PER_BLOCK + wid;
    const long long rowBase    = waveGlobal * ROWS_PER_WAVE;

    float acc = 0.0f;
    for (int r = 0; r < ROWS_PER_WAVE; ++r) {
        const float* rowp = inp + (rowBase + r) * (long long)Csz;

        // Prefetch next row: 32 lanes x 128B lines = 4KB (one full row).
        if (r + 1 < ROWS_PER_WAVE) {
            const char* np = (const char*)(rowp + Csz) + lane * 128;
            __builtin_prefetch(np, 0, 0);   // -> global_prefetch_b8
        }

        float m = -INFINITY;
        float s = 0.0f;

        // 31 uniform full-wave iterations: no bounds check, no EXEC masking.
        #pragma unroll 4
        for (int k = 0; k < FULL_ITERS; ++k) {
            float x = __builtin_nontemporal_load(rowp + k * WAVE + lane);
            online_update(x, m, s);
        }

        // Tail: columns 992..1000 live in lanes 0..8; lane 8 holds the last column.
        float xl = 0.0f;
        if (lane < TAIL_LANES) {
            float x = __builtin_nontemporal_load(rowp + TAIL_BASE + lane);
            online_update(x, m, s);
            if (lane == LAST_LANE) xl = x;
        }

        // wave32 butterfly combine of (m, s).
        #pragma unroll
        for (int off = WAVE >> 1; off > 0; off >>= 1) {
            float mo = __shfl_xor(m, off, WAVE);
            float so = __shfl_xor(s, off, WAVE);
            float nm = fmaxf(m, mo);
            s = s * expf(m - nm) + so * expf(mo - nm);
            m = nm;
        }
        float xlast = __shfl(xl, LAST_LANE, WAVE);

        float p = expf(xlast - m) / s;
        float w = (p == 1.0f) ? (1.0f - EPSC) : 1.0f;
        acc += logf(1.0f - p * w);          // same value in every lane
    }

    __shared__ float wsum[WAVES_PER_BLOCK];
    if (lane == 0) wsum[wid] = acc;
    __syncthreads();
    if (threadIdx.x == 0) {
        float t = 0.0f;
        #pragma unroll
        for (int i = 0; i < WAVES_PER_BLOCK; ++i) t += wsum[i];
        partial[blockIdx.x] = t;            // sum of 64 rows' losses
    }
}

// Stage 2: deterministic single-block reduction of the 4096 partials (fp64 accum).
__global__ __launch_bounds__(256)
void reduce_kernel(const float* __restrict__ partial, float* __restrict__ out) {
    __shared__ double sd[256];
    double t = 0.0;
    for (int i = threadIdx.x; i < GRID1; i += 256) t += (double)partial[i];
    sd[threadIdx.x] = t;
    __syncthreads();
    #pragma unroll
    for (int off = 128; off > 0; off >>= 1) {
        if (threadIdx.x < off) sd[threadIdx.x] += sd[threadIdx.x + off];
        __syncthreads();
    }
    if (threadIdx.x == 0) out[0] = (float)(sd[0] / (double)Bsz);
}

extern "C" void kernel_launch(void* const* d_in, const int* in_sizes, int n_in,
                              void* d_out, int out_size, void* d_ws, size_t ws_size,
                              hipStream_t stream) {
    const float* inp = (const float*)d_in[0];
    // d_in[1] (int64 target) is unused: the reference loss ignores it.
    float* partial = (float*)d_ws;     // GRID1 floats = 16 KB of scratch
    float* out     = (float*)d_out;

    row_loss_kernel<<<GRID1, THREADS, 0, stream>>>(inp, partial);
    reduce_kernel<<<1, 256, 0, stream>>>(partial, out);
}